// Apply_on_single_area_43104291782950
// MI455X (gfx1250) — compile-verified
//
#include <hip/hip_runtime.h>

typedef __attribute__((ext_vector_type(2))) float v2f;
typedef __attribute__((ext_vector_type(8))) float v8f;

#define DXY 32
#define NCH 4

// One wave32 per image. lane = column j. Row-axis erode in registers,
// column-axis erode via lane shuffles, final reduction via v_wmma_f32_16x16x4_f32.
__global__ __launch_bounds__(256) void erode_loss_kernel(
    const float* __restrict__ mask,   // [B,32,32,4]
    const float* __restrict__ em,     // [B,32,32,1]
    const int*   __restrict__ midx_p, // [1]
    float*       __restrict__ out,    // [B]
    int nB)
{
    const int lane = threadIdx.x & 31;
    const int wave = threadIdx.x >> 5;
    const int b    = blockIdx.x * 8 + wave;
    if (b >= nB) return;              // wave-uniform: EXEC stays all-ones

    const int mi = *midx_p;           // uniform scalar load

    // Channel selected via the address (mi is uniform) -> branchless b32 loads.
    // Lanes are 16B apart; a row's 32 lanes cover the same four 128B lines a
    // b128 load would, so HBM traffic is identical with zero select overhead.
    // Reference zeroes row 31 and column 31 before eroding: row 31 is never
    // loaded, lane 31 is forced to 0.
    const float* mcol = mask + (size_t)b * (DXY * DXY * NCH) + lane * NCH + mi;

    float m[DXY];
#pragma unroll
    for (int i = 0; i < DXY - 1; ++i) {
        float f = __builtin_nontemporal_load(mcol + i * (DXY * NCH));
        m[i] = (lane == DXY - 1) ? 0.0f : f;
    }
    m[DXY - 1] = 0.0f;

    const float* emb = em + (size_t)b * (DXY * DXY);

    // erode(a,b) = a*b + (1-a)*a + (1-b)*a with a = next*curr, b = curr*prev
    float partial = 0.0f;
#pragma unroll
    for (int i = 0; i < DXY - 1; ++i) {   // row 31: curr==0 -> eroded==0, skip
        float curr = m[i];

        // axis 1 (rows): neighbors live in this lane's registers
        float nR = m[i + 1];                       // m[31]==0 handles i==30
        float pR = (i > 0) ? m[i - 1] : 0.0f;      // zero padding above
        float ax = nR * curr;
        float bx = curr * pR;
        float ex = ax * bx + (1.0f - ax) * ax + (1.0f - bx) * ax;

        // axis 2 (cols): neighbors via wave32 shuffles, zero padding at edges
        float nC = __shfl_down(curr, 1, 32);
        if (lane == 31) nC = 0.0f;
        float pC = __shfl_up(curr, 1, 32);
        if (lane == 0) pC = 0.0f;
        float ay = nC * curr;
        float by = curr * pC;
        float ey = ay * by + (1.0f - ay) * ay + (1.0f - by) * ay;

        float e = __builtin_nontemporal_load(emb + i * DXY + lane);
        partial = fmaf(ex * ey, e, partial);
    }

    // Wave reduction via WMMA: A is 16x4 f32 (2 VGPRs). Lanes 0-15 carry
    // A[m][K0], lanes 16-31 carry A[m][K2]; K1/K3 are zero. With B == ones,
    // D[m][n] = partial[m] + partial[m+16] for every n. Lane 0 then holds
    // D[0..7][0] in c[0..7] and lane 16 holds D[8..15][0].
    v2f a;  a.x = partial; a.y = 0.0f;
    v2f bo; bo.x = 1.0f;   bo.y = 1.0f;
    v8f c = {0.0f, 0.0f, 0.0f, 0.0f, 0.0f, 0.0f, 0.0f, 0.0f};
    c = __builtin_amdgcn_wmma_f32_16x16x4_f32(
        /*neg_a=*/false, a, /*neg_b=*/false, bo,
        /*c_mod=*/(short)0, c, /*reuse_a=*/false, /*reuse_b=*/false);

    float s = c[0] + c[1] + c[2] + c[3] + c[4] + c[5] + c[6] + c[7];
    s += __shfl_down(s, 16, 32);          // lane0 += lane16
    if (lane == 0) out[b] = s;
}

extern "C" void kernel_launch(void* const* d_in, const int* in_sizes, int n_in,
                              void* d_out, int out_size, void* d_ws, size_t ws_size,
                              hipStream_t stream) {
    // d_in[0] = resized_image (dead in the reference -> never touched)
    const float* mask = (const float*)d_in[1];   // [B,32,32,4] f32
    const float* em   = (const float*)d_in[2];   // [B,32,32,1] f32
    const int*   midx = (const int*)d_in[3];     // [1] int
    float* out = (float*)d_out;                  // [B] f32

    const int B = in_sizes[1] / (DXY * DXY * NCH);
    const int blocks = (B + 7) / 8;              // 8 waves (images) per block
    erode_loss_kernel<<<blocks, 256, 0, stream>>>(mask, em, midx, out, B);
}